// JointAttention_84164179133006
// MI455X (gfx1250) — compile-verified
//
#include <hip/hip_runtime.h>

// ---------------------------------------------------------------------------
// JointAttention for MI455X (gfx1250, wave32, WMMA).
// B=64, N=15 (N1=16 kv streams), T=256, H=128, TEMP=1.
// f32 math throughout via V_WMMA_F32_16X16X4_F32.
// ---------------------------------------------------------------------------

#define BB 64
#define NN 15
#define N1 16
#define TT 256
#define HH 128
#define KV (N1 * TT)   // 4096 keys per query row

typedef __attribute__((ext_vector_type(2))) float v2f;
typedef __attribute__((ext_vector_type(8))) float v8f;

// D = A(16x4 f32) * B(4x16 f32) + C  -- wave32 matrix op
__device__ __forceinline__ v8f wmma_f32(v2f a, v2f b, v8f c) {
    return __builtin_amdgcn_wmma_f32_16x16x4_f32(
        /*neg_a=*/false, a, /*neg_b=*/false, b,
        /*c_mod=*/(short)0, c, /*reuse_a=*/false, /*reuse_b=*/false);
}

// ---------------------------------------------------------------------------
// Kernel 1: Y[m, n] = sum_h X[m, h] * W[n, h] + bias[n]
// kvMode==0 : rows = B*T, X row m -> node[m]
// kvMode==1 : rows = B*N1*T, X row (b, j, t) -> j==0 ? node[b,t] : neigh[b,j-1,t]
// Block: 256 threads = 8 waves; each wave owns one 16-wide n-tile (n0 = wave*16).
// Grid.x = rows / 16.
// ---------------------------------------------------------------------------
__global__ void proj_kernel(const float* __restrict__ node,
                            const float* __restrict__ neigh,
                            const float* __restrict__ W,
                            const float* __restrict__ bias,
                            float* __restrict__ Y,
                            int kvMode) {
    const int wave = threadIdx.x >> 5;
    const int lane = threadIdx.x & 31;
    const int lm = lane & 15;   // L % 16
    const int lh = lane >> 4;   // L / 16
    const int m0 = blockIdx.x * 16;
    const int n0 = wave * 16;

    // source row for A-fragment (lanes 16-31 reuse rows of lanes 0-15)
    const int m = m0 + lm;
    const float* xrow;
    if (kvMode) {
        int b = m / (N1 * TT);
        int r = m - b * (N1 * TT);
        int j = r / TT;
        int t = r - j * TT;
        xrow = (j == 0) ? (node + ((size_t)b * TT + t) * HH)
                        : (neigh + (((size_t)b * NN + (j - 1)) * TT + t) * HH);
    } else {
        xrow = node + (size_t)m * HH;
    }
    const float* wrow = W + (size_t)(n0 + lm) * HH;  // row n of W (B[k][n] = W[n][k])

    v8f acc = {};
#pragma unroll 8
    for (int k0 = 0; k0 < HH; k0 += 4) {
        const int ka = k0 + 2 * lh;                  // this lane's K base
        v2f afrag = *(const v2f*)(xrow + ka);        // A[m][ka], A[m][ka+1]
        v2f bfrag = *(const v2f*)(wrow + ka);        // B[ka][n], B[ka+1][n]
        acc = wmma_f32(afrag, bfrag, acc);
    }

    const float bval = bias[n0 + lm];
#pragma unroll
    for (int r = 0; r < 8; ++r) {
        const int mm = m0 + r + 8 * lh;              // D vgpr r -> row r + 8*(L/16)
        Y[(size_t)mm * HH + (n0 + lm)] = acc[r] + bval;
    }
}

// ---------------------------------------------------------------------------
// Kernel 2: S[b,t,j,s] = sum_h q[b,t,h] * k[b,j,s,h], masked (s>t -> -inf),
// plus pre-mask Frobenius accumulation into snormAcc[b].
// Grid.x decomposes as (b, j, t-tile, s-half); 8 waves cover 8 s-tiles of 16.
// ---------------------------------------------------------------------------
__global__ void score_kernel(const float* __restrict__ q,
                             const float* __restrict__ k,
                             float* __restrict__ S,        // [B, T, N1, T]
                             float* __restrict__ snormAcc) // [B]
{
    int id = blockIdx.x;
    const int sh = id & 1;  id >>= 1;   // s half (0 / 1)
    const int tt = id & 15; id >>= 4;   // t tile
    const int j  = id & 15; id >>= 4;   // kv stream
    const int b  = id;

    const int wave = threadIdx.x >> 5;
    const int lane = threadIdx.x & 31;
    const int lm = lane & 15;
    const int lh = lane >> 4;
    const int t0 = tt * 16;
    const int s0 = sh * 128 + wave * 16;

    const float* qrow = q + ((size_t)b * TT + (t0 + lm)) * HH;
    const float* krow = k + (((size_t)b * N1 + j) * TT + (s0 + lm)) * HH;

    v8f acc = {};
#pragma unroll 8
    for (int k0 = 0; k0 < HH; k0 += 4) {
        const int ka = k0 + 2 * lh;
        v2f afrag = *(const v2f*)(qrow + ka);        // q[t][h]
        v2f bfrag = *(const v2f*)(krow + ka);        // B[h][s] = k[s][h]
        acc = wmma_f32(afrag, bfrag, acc);
    }

    // pre-mask sum of squares -> wave reduce -> one atomic per wave
    float ss = 0.0f;
#pragma unroll
    for (int r = 0; r < 8; ++r) ss += acc[r] * acc[r];
#pragma unroll
    for (int off = 16; off > 0; off >>= 1) ss += __shfl_xor(ss, off, 32);
    if (lane == 0) atomicAdd(&snormAcc[b], ss);

    // masked store in [b, t, j, s] order (TEMP == 1)
    const int s = s0 + lm;
#pragma unroll
    for (int r = 0; r < 8; ++r) {
        const int t = t0 + r + 8 * lh;
        const float val = (s > t) ? -__builtin_inff() : acc[r];
        S[(((size_t)b * TT + t) * N1 + j) * TT + s] = val;
    }
}

// ---------------------------------------------------------------------------
// Kernel 3: in-place softmax over each contiguous row of KV=4096 scores.
// One 256-thread block per (b,t) row; each thread strides 16 elements.
// ---------------------------------------------------------------------------
__global__ void softmax_kernel(float* __restrict__ A) {
    float* p = A + (size_t)blockIdx.x * KV;
    const int tid = threadIdx.x;
    const int wave = tid >> 5;
    const int lane = tid & 31;
    __shared__ float red[8];

    float vals[16];
    float mx = -__builtin_inff();
#pragma unroll
    for (int it = 0; it < 16; ++it) {
        const float x = p[tid + it * 256];
        vals[it] = x;
        mx = fmaxf(mx, x);
    }
#pragma unroll
    for (int off = 16; off > 0; off >>= 1) mx = fmaxf(mx, __shfl_xor(mx, off, 32));
    if (lane == 0) red[wave] = mx;
    __syncthreads();
    float m = red[0];
#pragma unroll
    for (int w = 1; w < 8; ++w) m = fmaxf(m, red[w]);
    __syncthreads();

    float sum = 0.0f;
#pragma unroll
    for (int it = 0; it < 16; ++it) {
        const float e = __expf(vals[it] - m);   // exp(-inf) == 0 for masked keys
        vals[it] = e;
        sum += e;
    }
#pragma unroll
    for (int off = 16; off > 0; off >>= 1) sum += __shfl_xor(sum, off, 32);
    if (lane == 0) red[wave] = sum;
    __syncthreads();
    float tot = 0.0f;
#pragma unroll
    for (int w = 0; w < 8; ++w) tot += red[w];
    const float inv = 1.0f / tot;
#pragma unroll
    for (int it = 0; it < 16; ++it) p[tid + it * 256] = vals[it] * inv;
}

// ---------------------------------------------------------------------------
// Kernel 4: out[b,t,n] = sum_kk A[b,t,kk] * v[b,kk,n], K = 4096.
// Grid.x = B * (T/16); 8 waves cover the 8 n-tiles of H=128.
// ---------------------------------------------------------------------------
__global__ void out_kernel(const float* __restrict__ A,
                           const float* __restrict__ v,
                           float* __restrict__ out) {
    int id = blockIdx.x;
    const int tt = id & 15;
    const int b  = id >> 4;
    const int wave = threadIdx.x >> 5;
    const int lane = threadIdx.x & 31;
    const int lm = lane & 15;
    const int lh = lane >> 4;
    const int t0 = tt * 16;
    const int n0 = wave * 16;

    const float* arow  = A + ((size_t)b * TT + (t0 + lm)) * KV;
    const float* vbase = v + (size_t)b * KV * HH + (n0 + lm);

    v8f acc = {};
#pragma unroll 4
    for (int k0 = 0; k0 < KV; k0 += 4) {
        const int ka = k0 + 2 * lh;
        v2f afrag = *(const v2f*)(arow + ka);        // A[t][ka], A[t][ka+1]
        v2f bfrag;
        bfrag.x = vbase[(size_t)ka * HH];            // v[ka][n]
        bfrag.y = vbase[(size_t)(ka + 1) * HH];      // v[ka+1][n]
        acc = wmma_f32(afrag, bfrag, acc);
    }
#pragma unroll
    for (int r = 0; r < 8; ++r) {
        const int t = t0 + r + 8 * lh;
        out[((size_t)b * TT + t) * HH + (n0 + lm)] = acc[r];
    }
}

// ---------------------------------------------------------------------------
// Kernel 5: S_norm[b] = sqrt(acc[b])
// ---------------------------------------------------------------------------
__global__ void snorm_kernel(const float* __restrict__ acc,
                             float* __restrict__ snorm) {
    const int b = threadIdx.x;
    if (b < BB) snorm[b] = sqrtf(acc[b]);
}

// ---------------------------------------------------------------------------
extern "C" void kernel_launch(void* const* d_in, const int* in_sizes, int n_in,
                              void* d_out, int out_size, void* d_ws, size_t ws_size,
                              hipStream_t stream) {
    (void)in_sizes; (void)n_in; (void)out_size; (void)ws_size;

    const float* node  = (const float*)d_in[0];   // [B, T, H]
    const float* neigh = (const float*)d_in[1];   // [B, N, T, H]
    // d_in[2] = neighbors_number (15, fixed), d_in[3] = causal mask (recomputed analytically)
    const float* Wq = (const float*)d_in[4];
    const float* bq = (const float*)d_in[5];
    const float* Wk = (const float*)d_in[6];
    const float* bk = (const float*)d_in[7];
    const float* Wv = (const float*)d_in[8];
    const float* bv = (const float*)d_in[9];

    float* out   = (float*)d_out;                         // [B, T, H]
    float* Amat  = out + (size_t)BB * TT * HH;            // [B, T, N1, T]
    float* Snorm = Amat + (size_t)BB * TT * N1 * TT;      // [B]

    float* q  = (float*)d_ws;                             // [B*T, H]
    float* k  = q + (size_t)BB * TT * HH;                 // [B*N1*T, H]
    float* v  = k + (size_t)BB * N1 * TT * HH;            // [B*N1*T, H]
    float* sn = v + (size_t)BB * N1 * TT * HH;            // [B] accumulator

    hipMemsetAsync(sn, 0, BB * sizeof(float), stream);

    // projections
    proj_kernel<<<(BB * TT) / 16,      256, 0, stream>>>(node, neigh, Wq, bq, q, 0);
    proj_kernel<<<(BB * N1 * TT) / 16, 256, 0, stream>>>(node, neigh, Wk, bk, k, 1);
    proj_kernel<<<(BB * N1 * TT) / 16, 256, 0, stream>>>(node, neigh, Wv, bv, v, 1);

    // raw masked scores + pre-mask Frobenius accumulation
    score_kernel<<<BB * N1 * (TT / 16) * (TT / 128), 256, 0, stream>>>(q, k, Amat, sn);

    // joint softmax over all 4096 keys per (b,t)
    softmax_kernel<<<BB * TT, 256, 0, stream>>>(Amat);

    // attention output
    out_kernel<<<BB * (TT / 16), 256, 0, stream>>>(Amat, v, out);

    // S_norm tail
    snorm_kernel<<<1, 64, 0, stream>>>(sn, Snorm);
}